// Attention_18760417149684
// MI455X (gfx1250) — compile-verified
//
#include <hip/hip_runtime.h>
#include <hip/hip_bf16.h>
#include <cstdint>

// Problem constants (from reference): N=64, T=2048, D=256, fp32.
#define N_ 64
#define T_ 2048
#define D_ 256
#define NCHUNK 8    // t-split for the context pass (partials in d_ws)
#define KCH 64      // key floats per staged chunk (per row)
#define KCHP 68     // padded LDS row stride (floats) -> bank-conflict-free

typedef __attribute__((ext_vector_type(2))) float v2f;
typedef __attribute__((ext_vector_type(8))) float v8f;

// ---------------------------------------------------------------------------
// Kernel 1: energy[n,t] = <key[n,t,:], query[n,:]> via V_WMMA_F32_16X16X4_F32.
// A-fragment = query chunk broadcast across the 16 rows; B-fragment = 4x16 key
// tile (columns = 16 consecutive t). Every row of the 16x16 result equals the
// 16 energies, so lanes 0..15 read VGPR0 (row M=0, N=lane) and store.
//
// Key data is streamed to LDS with global_load_async_to_lds_b128 (fully
// coalesced: 2 rows x 16 lanes x 16B = 512B per instruction, as two contiguous 256B runs),
// double-buffered per wave, synchronized with s_wait_asynccnt. WMMA B
// fragments are then conflict-free ds reads (padded row stride).
//
// Grid: (N, T/128); block 256 = 8 waves, one 16-t tile per wave.
// Energies land in the norm_attention region of d_out (softmax overwrites
// in place).
// ---------------------------------------------------------------------------
__global__ __launch_bounds__(256) void attn_energy_wmma(
    const float* __restrict__ query,   // [N, D]
    const float* __restrict__ key,     // [N, T, D]
    float* __restrict__ att)           // [N, T] (energy out)
{
    const int n    = blockIdx.x;
    const int t0   = blockIdx.y * 128;      // 8 waves * 16 rows
    const int tid  = threadIdx.x;
    const int wave = tid >> 5;
    const int lane = tid & 31;

    __shared__ float qs[D_];
    __shared__ float kbuf[8][2][16 * KCHP];  // [wave][buf][row*KCHP + col]

    qs[tid] = query[n * D_ + tid];
    __syncthreads();

    const int tw  = t0 + wave * 16;         // this wave's 16 t-rows
    const int off = (lane >= 16) ? 2 : 0;   // K=0,1 vs K=2,3 sub-rows

    const float* kbase = key + ((size_t)n * T_ + (size_t)tw) * D_;
    const uint32_t ldsb0 = (uint32_t)(uintptr_t)&kbuf[wave][0][0];
    const uint32_t ldsb1 = (uint32_t)(uintptr_t)&kbuf[wave][1][0];

    // Coalesced async stage of one 16-row x 64-float chunk into LDS.
    // Instruction j covers rows {2j, 2j+1}: lane -> (row = 2j + lane/16,
    // col4 = lane%16), i.e. two contiguous 256-byte row runs per issue.
    auto stage = [&](int c, uint32_t dstbase) {
        const float* src = kbase + (size_t)(lane >> 4) * D_
                         + (size_t)c * KCH + (size_t)(lane & 15) * 4;
        uint32_t dst = dstbase
                     + (uint32_t)(((lane >> 4) * KCHP + (lane & 15) * 4) * 4);
        #pragma unroll
        for (int j = 0; j < 8; ++j) {
            asm volatile("global_load_async_to_lds_b128 %0, %1, off"
                         :: "v"(dst), "v"(src) : "memory");
            src += 2 * D_;
            dst += 2 * KCHP * 4;
        }
    };

    stage(0, ldsb0);
    asm volatile("s_wait_asynccnt 0x0" ::: "memory");

    v8f acc = {0.f, 0.f, 0.f, 0.f, 0.f, 0.f, 0.f, 0.f};
    #pragma unroll
    for (int c = 0; c < D_ / KCH; ++c) {           // 4 chunks
        if (c + 1 < D_ / KCH)
            stage(c + 1, ((c + 1) & 1) ? ldsb1 : ldsb0);

        const float* kb = &kbuf[wave][c & 1][(lane & 15) * KCHP];
        #pragma unroll 4
        for (int d0 = 0; d0 < KCH; d0 += 4) {
            v2f a = *(const v2f*)(&qs[c * KCH + d0 + off]); // broadcast read
            v2f b = *(const v2f*)(&kb[d0 + off]);           // B[k,n] fragment
            // (neg_a, A, neg_b, B, c_mod, C, reuse_a, reuse_b)
            acc = __builtin_amdgcn_wmma_f32_16x16x4_f32(false, a, false, b,
                                                        (short)0, acc,
                                                        false, false);
        }
        if (c + 1 < D_ / KCH)
            asm volatile("s_wait_asynccnt 0x0" ::: "memory");
    }
    // All result rows identical; row M=0 is VGPR0 of lanes 0..15 (N = lane).
    if (lane < 16) att[(size_t)n * T_ + (size_t)(tw + lane)] = acc[0];
}

// ---------------------------------------------------------------------------
// Kernel 2: masked softmax with renormalization, in place on att[n,:].
// softmax-over-T + mask + renorm == softmax over t < len[n]; zeros elsewhere.
// Grid: N blocks of 256 threads.
// ---------------------------------------------------------------------------
__global__ __launch_bounds__(256) void attn_softmax(
    float* __restrict__ att,           // [N, T] energy in / attention out
    const int* __restrict__ lens)      // [N]
{
    const int n   = blockIdx.x;
    const int tid = threadIdx.x;
    const int len = lens[n];
    float* e = att + (size_t)n * T_;

    __shared__ float red[256];

    float m = -__builtin_inff();
    for (int t = tid; t < len; t += 256) m = fmaxf(m, e[t]);
    red[tid] = m;
    __syncthreads();
    for (int s = 128; s > 0; s >>= 1) {
        if (tid < s) red[tid] = fmaxf(red[tid], red[tid + s]);
        __syncthreads();
    }
    m = red[0];
    __syncthreads();

    float sum = 0.f;
    for (int t = tid; t < len; t += 256) sum += __expf(e[t] - m);
    red[tid] = sum;
    __syncthreads();
    for (int s = 128; s > 0; s >>= 1) {
        if (tid < s) red[tid] += red[tid + s];
        __syncthreads();
    }
    const float inv = 1.0f / red[0];

    for (int t = tid; t < T_; t += 256) {
        float v = (t < len) ? __expf(e[t] - m) * inv : 0.0f;
        e[t] = v;
    }
}

// ---------------------------------------------------------------------------
// Kernel 3: partial context. context[n,d] = sum_t att[n,t]*value[n,t,d].
// T split into NCHUNK chunks; each WG streams its chunk of V with coalesced
// float4 (b128) loads, 4 t-rows in flight, early exit at len[n] (att is zero
// beyond it -> saves ~half the V bandwidth on average).
// Grid: (N, NCHUNK); block 256.
// ---------------------------------------------------------------------------
__global__ __launch_bounds__(256) void attn_context_partial(
    const float* __restrict__ att,     // [N, T]
    const float* __restrict__ value,   // [N, T, D]
    const int* __restrict__ lens,      // [N]
    float* __restrict__ partial)       // [N, NCHUNK, D]
{
    const int n     = blockIdx.x;
    const int chunk = blockIdx.y;
    const int tid   = threadIdx.x;
    const int d4    = (tid & 63) * 4;  // float4 column
    const int tsub  = tid >> 6;        // 0..3 interleaved t streams

    const int tstart = chunk * (T_ / NCHUNK);
    const int len    = lens[n];
    const int tend   = min(tstart + T_ / NCHUNK, len);

    const float* vp = value + (size_t)n * T_ * D_;
    const float* ap = att + (size_t)n * T_;

    float4 acc = make_float4(0.f, 0.f, 0.f, 0.f);
    for (int t = tstart + tsub; t < tend; t += 4) {
        if (t + 8 < tend)
            __builtin_prefetch(vp + (size_t)(t + 8) * D_ + d4, 0, 1);
        const float  w = ap[t];
        const float4 v = *(const float4*)(vp + (size_t)t * D_ + d4);
        acc.x += w * v.x;
        acc.y += w * v.y;
        acc.z += w * v.z;
        acc.w += w * v.w;
    }

    __shared__ float4 red[256];
    red[tid] = acc;
    __syncthreads();
    if (tsub == 0) {
        float4 a = red[tid];
        #pragma unroll
        for (int s = 1; s < 4; ++s) {
            float4 b = red[tid + 64 * s];
            a.x += b.x; a.y += b.y; a.z += b.z; a.w += b.w;
        }
        float* out = partial + ((size_t)(n * NCHUNK + chunk)) * D_ + d4;
        *(float4*)out = a;
    }
}

// ---------------------------------------------------------------------------
// Kernel 4: reduce NCHUNK partials into context. Grid: N*D/256 blocks of 256.
// ---------------------------------------------------------------------------
__global__ __launch_bounds__(256) void attn_context_reduce(
    const float* __restrict__ partial, // [N, NCHUNK, D]
    float* __restrict__ ctx)           // [N, D]
{
    const int idx = blockIdx.x * 256 + threadIdx.x;  // over N*D
    const int n = idx / D_;
    const int d = idx % D_;
    float s = 0.f;
    #pragma unroll
    for (int c = 0; c < NCHUNK; ++c)
        s += partial[((size_t)(n * NCHUNK + c)) * D_ + d];
    ctx[idx] = s;
}

// ---------------------------------------------------------------------------
// Launch. Inputs: query f32 [N,D], key f32 [N,T,D], value f32 [N,T,D],
// lens int [N]. Output: context f32 [N,D] then norm_attention f32 [N,T],
// concatenated flat. Workspace: N*NCHUNK*D floats (512 KB) of partials.
// ---------------------------------------------------------------------------
extern "C" void kernel_launch(void* const* d_in, const int* in_sizes, int n_in,
                              void* d_out, int out_size, void* d_ws, size_t ws_size,
                              hipStream_t stream) {
    const float* query = (const float*)d_in[0];
    const float* key   = (const float*)d_in[1];
    const float* value = (const float*)d_in[2];
    const int*   lens  = (const int*)d_in[3];

    float* ctx = (float*)d_out;             // [N, D]
    float* att = (float*)d_out + N_ * D_;   // [N, T]
    float* partial = (float*)d_ws;          // [N, NCHUNK, D]

    attn_energy_wmma<<<dim3(N_, T_ / 128), 256, 0, stream>>>(query, key, att);
    attn_softmax<<<N_, 256, 0, stream>>>(att, lens);
    attn_context_partial<<<dim3(N_, NCHUNK), 256, 0, stream>>>(att, value, lens, partial);
    attn_context_reduce<<<(N_ * D_) / 256, 256, 0, stream>>>(partial, ctx);
}